// GCNRegression_77833397338747
// MI455X (gfx1250) — compile-verified
//
#include <hip/hip_runtime.h>

typedef __attribute__((ext_vector_type(2))) float v2f;
typedef __attribute__((ext_vector_type(8))) float v8f;

#define IN_CH 128
#define HID   64

// ---------------------------------------------------------------- degree init
__global__ void __launch_bounds__(256)
deg_init_kernel(float* __restrict__ deg, int N) {
    int i = blockIdx.x * 256 + threadIdx.x;
    if (i < N) deg[i] = 1.0f;  // self-loop contributes 1 to its own dst
}

// ------------------------------------------------------------- degree count
__global__ void __launch_bounds__(256)
deg_count_kernel(const int* __restrict__ ei, float* __restrict__ deg, long long E) {
    long long e = (long long)blockIdx.x * 256 + threadIdx.x;
    if (e < E) {
        int dst = ei[E + e];               // edge_index[1][e]
        unsafeAtomicAdd(&deg[dst], 1.0f);  // native global_atomic_add_f32
    }
}

// --------------------------------------------------------- dinv = rsqrt(deg)
__global__ void __launch_bounds__(256)
dinv_kernel(float* __restrict__ deg, int N) {
    int i = blockIdx.x * 256 + threadIdx.x;
    if (i < N) {
        float d = deg[i];
        deg[i] = (d > 0.0f) ? rsqrtf(d) : 0.0f;
    }
}

// --------------------------------------------------------------- WMMA GEMM
// hs[n][:] = (x[n][:] @ W1^T) * dinv[n]    (row-scaled projection)
// agg initialized to hs (the self-loop term of the scatter sum).
// One block = 16 rows of x; 4 waves, each owning one 16-wide N-tile of HID=64.
// fp32 WMMA 16x16x4: A frag = 2 VGPRs (lanes 0-15: K+0/K+1, lanes 16-31: K+2/K+3),
// B frag mirrored with N on lanes; C/D v8f: VGPR r -> rows {r, r+8}.
__global__ void __launch_bounds__(128)
gemm_wmma_kernel(const float* __restrict__ x, const float* __restrict__ W1,
                 const float* __restrict__ dinv, float* __restrict__ hs,
                 float* __restrict__ agg, int N) {
    const int lane = threadIdx.x & 31;
    const int wave = threadIdx.x >> 5;   // 0..3 -> N-tile
    const int half = lane >> 4;          // 0: K+0/1, 1: K+2/3
    const int lr   = lane & 15;
    const int m0   = blockIdx.x * 16;
    const int n0   = wave * 16;

    int mrow = m0 + lr;
    if (mrow >= N) mrow = N - 1;         // clamp (N is a multiple of 16 anyway)
    const float* xrow = x  + (long long)mrow      * IN_CH;
    const float* wrow = W1 + (long long)(n0 + lr) * IN_CH;  // W1 is [HID][IN_CH]

    v8f c = {};
#pragma unroll
    for (int k = 0; k < IN_CH; k += 4) {
        const int kk = k + half * 2;
        v2f a = *(const v2f*)(xrow + kk);   // A[m][kk..kk+1]
        v2f b = *(const v2f*)(wrow + kk);   // B[kk..kk+1][n] == W1[n][kk..kk+1]
        c = __builtin_amdgcn_wmma_f32_16x16x4_f32(
                /*neg_a=*/false, a, /*neg_b=*/false, b,
                /*c_mod=*/(short)0, c, /*reuse_a=*/false, /*reuse_b=*/false);
    }

#pragma unroll
    for (int r = 0; r < 8; ++r) {
        const int m = m0 + half * 8 + r;   // VGPR r holds rows r (lo half) / r+8 (hi half)
        if (m < N) {
            const int n = n0 + lr;
            const float v = c[r] * dinv[m];
            hs[(long long)m * HID + n]  = v;
            agg[(long long)m * HID + n] = v;   // self-loop term
        }
    }
}

// ------------------------------------------------------------- edge scatter
// agg[dst][:] += hs[src][:].  One wave per edge, float2 per lane (64 channels).
// hs & agg (2 x 25.6 MB) are L2-resident on MI455X (192 MB L2).
__global__ void __launch_bounds__(256)
scatter_kernel(const int* __restrict__ ei, const float* __restrict__ hs,
               float* __restrict__ agg, long long E) {
    long long tid = (long long)blockIdx.x * 256 + threadIdx.x;
    long long e = tid >> 5;
    if (e >= E) return;
    const int pair = (int)(tid & 31) * 2;
    const int src = ei[e];        // edge_index[0][e]  (wave-uniform -> scalar load)
    const int dst = ei[E + e];    // edge_index[1][e]
    const v2f v = *(const v2f*)(hs + (long long)src * HID + pair);
    float* p = agg + (long long)dst * HID + pair;
    unsafeAtomicAdd(p,     v.x);
    unsafeAtomicAdd(p + 1, v.y);
}

// ---------------------------------------------------------------- finalize
// out[n] = relu(dinv[n]*agg[n][:] + b1) . W2 + b2     (wave per node)
__global__ void __launch_bounds__(256)
finalize_kernel(const float* __restrict__ agg, const float* __restrict__ dinv,
                const float* __restrict__ b1, const float* __restrict__ W2,
                const float* __restrict__ b2, float* __restrict__ out, int N) {
    const int wave = threadIdx.x >> 5;
    const int lane = threadIdx.x & 31;
    const int node = blockIdx.x * 8 + wave;
    if (node >= N) return;
    const float dv = dinv[node];
    const float* row = agg + (long long)node * HID;
    float acc = 0.0f;
#pragma unroll
    for (int i = 0; i < 2; ++i) {
        const int ch = lane + i * 32;
        const float t = fmaxf(dv * row[ch] + b1[ch], 0.0f);
        acc += t * W2[ch];
    }
#pragma unroll
    for (int off = 16; off > 0; off >>= 1)
        acc += __shfl_xor(acc, off, 32);
    if (lane == 0) out[node] = acc + b2[0];
}

// ---------------------------------------------------------------------------
extern "C" void kernel_launch(void* const* d_in, const int* in_sizes, int n_in,
                              void* d_out, int out_size, void* d_ws, size_t ws_size,
                              hipStream_t stream) {
    const float* x  = (const float*)d_in[0];
    const int*   ei = (const int*)d_in[1];   // int32 [2, E]
    const float* W1 = (const float*)d_in[2]; // [HID, IN_CH]
    const float* b1 = (const float*)d_in[3]; // [HID]
    const float* W2 = (const float*)d_in[4]; // [1, HID]
    const float* b2 = (const float*)d_in[5]; // [1]
    float* out = (float*)d_out;

    const int N = in_sizes[0] / IN_CH;
    const long long E = (long long)in_sizes[1] / 2;

    // workspace layout: deg/dinv [N] | hs [N*HID] | agg [N*HID]  (~51.6 MB)
    float* deg = (float*)d_ws;
    float* hs  = deg + N;
    float* agg = hs + (size_t)N * HID;

    deg_init_kernel<<<(N + 255) / 256, 256, 0, stream>>>(deg, N);
    deg_count_kernel<<<(unsigned)((E + 255) / 256), 256, 0, stream>>>(ei, deg, E);
    dinv_kernel<<<(N + 255) / 256, 256, 0, stream>>>(deg, N);
    gemm_wmma_kernel<<<(N + 15) / 16, 128, 0, stream>>>(x, W1, deg, hs, agg, N);
    const long long sthreads = E * 32;
    scatter_kernel<<<(unsigned)((sthreads + 255) / 256), 256, 0, stream>>>(ei, hs, agg, E);
    finalize_kernel<<<(N + 7) / 8, 256, 0, stream>>>(agg, deg, b1, W2, b2, out, N);
    (void)n_in; (void)out_size; (void)ws_size;
}